// DeQuPoolXYZ_6794638263040
// MI455X (gfx1250) — compile-verified
//
#include <hip/hip_runtime.h>
#include <math.h>

typedef float v2f __attribute__((ext_vector_type(2)));
typedef float v8f __attribute__((ext_vector_type(8)));
typedef unsigned int u32x4 __attribute__((ext_vector_type(4)));
typedef int i32x4 __attribute__((ext_vector_type(4)));
typedef int i32x8 __attribute__((ext_vector_type(8)));

#define DIM   4096
#define PLANE (DIM * DIM)   // 16,777,216 floats = 64 MB

#if __has_builtin(__builtin_amdgcn_tensor_load_to_lds) && __has_builtin(__builtin_amdgcn_s_wait_tensorcnt)
#define QPOOL_HAVE_TDM 1
#else
#define QPOOL_HAVE_TDM 0
#endif

// ---------------------------------------------------------------------------
// Complex helpers (device)
// ---------------------------------------------------------------------------
struct c32 { float x, y; };
__device__ inline c32 cmul(c32 a, c32 b) { return { a.x*b.x - a.y*b.y, a.x*b.y + a.y*b.x }; }
__device__ inline c32 cadd(c32 a, c32 b) { return { a.x + b.x, a.y + b.y }; }

__device__ inline void mat4_ident(c32 M[4][4]) {
    for (int i = 0; i < 4; ++i)
        for (int j = 0; j < 4; ++j)
            M[i][j] = { (i == j) ? 1.f : 0.f, 0.f };
}
__device__ inline void mat4_mul(c32 O[4][4], const c32 A[4][4], const c32 B[4][4]) {
    for (int i = 0; i < 4; ++i)
        for (int j = 0; j < 4; ++j) {
            c32 s = {0.f, 0.f};
            for (int k = 0; k < 4; ++k) s = cadd(s, cmul(A[i][k], B[k][j]));
            O[i][j] = s;
        }
}
__device__ inline void mat4_zero(c32 M[4][4]) {
    for (int i = 0; i < 4; ++i)
        for (int j = 0; j < 4; ++j) M[i][j] = {0.f, 0.f};
}
__device__ inline void kron_hi(c32 G[4][4], const c32 U[2][2]) {   // kron(U2, I2)
    mat4_zero(G);
    for (int i = 0; i < 2; ++i)
        for (int j = 0; j < 2; ++j)
            for (int k = 0; k < 2; ++k)
                G[2*i + k][2*j + k] = U[i][j];
}
__device__ inline void kron_lo(c32 G[4][4], const c32 U[2][2]) {   // kron(I2, U2)
    mat4_zero(G);
    for (int i = 0; i < 2; ++i)
        for (int k = 0; k < 2; ++k)
            for (int l = 0; l < 2; ++l)
                G[2*i + k][2*i + l] = U[k][l];
}
__device__ inline void make_rx(c32 U[2][2], float th) {
    float c = cosf(0.5f*th), s = sinf(0.5f*th);
    U[0][0] = {c, 0.f}; U[0][1] = {0.f, -s};
    U[1][0] = {0.f, -s}; U[1][1] = {c, 0.f};
}
__device__ inline void make_ry(c32 U[2][2], float th) {
    float c = cosf(0.5f*th), s = sinf(0.5f*th);
    U[0][0] = {c, 0.f}; U[0][1] = {-s, 0.f};
    U[1][0] = {s, 0.f}; U[1][1] = {c, 0.f};
}
__device__ inline void make_rz(c32 U[2][2], float th) {
    float c = cosf(0.5f*th), s = sinf(0.5f*th);   // e^{-i th/2} = c - i s
    U[0][0] = {c, -s}; U[0][1] = {0.f, 0.f};
    U[1][0] = {0.f, 0.f}; U[1][1] = {c, s};
}

// ---------------------------------------------------------------------------
// K0: build A = (V^dagger)^{kron 3}  (64x64 complex), single thread
// ---------------------------------------------------------------------------
__global__ void build_factor_kernel(const float* __restrict__ w,
                                    float* __restrict__ Are,
                                    float* __restrict__ Aim)
{
    if (threadIdx.x != 0 || blockIdx.x != 0) return;

    const float WM = 0.6324555320336759f;   // sqrt(2)*5^{-1/2}
    float t[6];
    for (int i = 0; i < 6; ++i) t[i] = w[i] * WM;

    c32 V[4][4], G[4][4], T[4][4], U2[2][2];
    mat4_ident(V);

#define APPLY() do { mat4_mul(T, G, V); \
    for (int _i = 0; _i < 4; ++_i) for (int _j = 0; _j < 4; ++_j) V[_i][_j] = T[_i][_j]; } while (0)

    make_rx(U2,  t[0]); kron_hi(G, U2); APPLY();
    make_rx(U2,  t[1]); kron_lo(G, U2); APPLY();
    make_ry(U2,  t[2]); kron_hi(G, U2); APPLY();
    make_ry(U2,  t[3]); kron_lo(G, U2); APPLY();
    make_rz(U2,  t[4]); kron_hi(G, U2); APPLY();
    make_rz(U2,  t[5]); kron_lo(G, U2); APPLY();
    mat4_zero(G);       // CNOT, control = high qubit
    G[0][0] = {1.f,0.f}; G[1][1] = {1.f,0.f}; G[2][3] = {1.f,0.f}; G[3][2] = {1.f,0.f};
    APPLY();
    make_rz(U2, -t[5]); kron_lo(G, U2); APPLY();
    make_ry(U2, -t[3]); kron_lo(G, U2); APPLY();
    make_rx(U2, -t[1]); kron_lo(G, U2); APPLY();
#undef APPLY

    c32 W[4][4];
    for (int i = 0; i < 4; ++i)
        for (int j = 0; j < 4; ++j)
            W[i][j] = { V[j][i].x, -V[j][i].y };

    for (int r = 0; r < 64; ++r)
        for (int c = 0; c < 64; ++c) {
            c32 p = cmul(W[r >> 4][c >> 4], W[(r >> 2) & 3][(c >> 2) & 3]);
            p = cmul(p, W[r & 3][c & 3]);
            Are[r*64 + c] = p.x;
            Aim[r*64 + c] = p.y;
        }
}

// ---------------------------------------------------------------------------
// Transpose kernels (LDS-tiled, coalesced both sides)
// ---------------------------------------------------------------------------
__global__ __launch_bounds__(256)
void transpose_real_kernel(const float* __restrict__ in, float* __restrict__ out)
{
    __shared__ float tile[32][33];
    const int bx = blockIdx.x * 32, by = blockIdx.y * 32;
    const int x = threadIdx.x & 31;
    const int y = threadIdx.x >> 5;   // 0..7
#pragma unroll
    for (int r = 0; r < 4; ++r)
        tile[y + 8*r][x] = in[(by + y + 8*r) * DIM + bx + x];
    __syncthreads();
#pragma unroll
    for (int r = 0; r < 4; ++r)
        out[(bx + y + 8*r) * DIM + by + x] = tile[x][y + 8*r];
}

__global__ __launch_bounds__(256)
void transpose_conj_kernel(const float* __restrict__ in_re, const float* __restrict__ in_im,
                           float* __restrict__ out_re, float* __restrict__ out_im)
{
    __shared__ float tre[32][33];
    __shared__ float tim[32][33];
    const int bx = blockIdx.x * 32, by = blockIdx.y * 32;
    const int x = threadIdx.x & 31;
    const int y = threadIdx.x >> 5;
#pragma unroll
    for (int r = 0; r < 4; ++r) {
        const int idx = (by + y + 8*r) * DIM + bx + x;
        tre[y + 8*r][x] = in_re[idx];
        tim[y + 8*r][x] = in_im[idx];
    }
    __syncthreads();
#pragma unroll
    for (int r = 0; r < 4; ++r) {
        const int idx = (bx + y + 8*r) * DIM + by + x;
        out_re[idx] =  tre[x][y + 8*r];
        out_im[idx] = -tim[x][y + 8*r];   // conjugate
    }
}

// ---------------------------------------------------------------------------
// WMMA wrapper: D = A(16x4 f32) x B(4x16 f32) + C(16x16 f32)
// ---------------------------------------------------------------------------
__device__ inline v8f cwmma(v2f a, v2f b, v8f c)
{
    return __builtin_amdgcn_wmma_f32_16x16x4_f32(false, a, false, b,
                                                 (short)0, c, false, false);
}

#if QPOOL_HAVE_TDM
// Issue one 2D TDM tile load: 64 rows x 128 floats, row stride = stride_k
// floats, into LDS at lds_off with 4-DWORD padding every 128 DWORDs.
__device__ inline void tdm_load_slab(const float* gbase, unsigned lds_off,
                                     int stride_k)
{
    unsigned long long ga = (unsigned long long)(uintptr_t)gbase;

    u32x4 g0;
    g0[0] = 1u;                                   // count=1, user descriptor
    g0[1] = lds_off;                              // LDS byte address
    g0[2] = (unsigned)(ga & 0xFFFFFFFFull);       // global addr [31:0]
    g0[3] = (unsigned)((ga >> 32) & 0x01FFFFFFull) | 0x80000000u; // [56:32] | type=2

    const unsigned TD = 0x400000u;                // large tensor dims (no clip)
    i32x8 g1;
    g1[0] = (int)((2u << 16)                      // data_size = 4 bytes
                | (1u << 20)                      // pad_enable
                | (6u << 22)                      // pad_interval = 128 DWORDs
                | (3u << 25));                    // pad_amount   = 4 DWORDs
    g1[1] = (int)((TD & 0xFFFFu) << 16);          // tensor_dim0 lo
    g1[2] = (int)((TD >> 16) | ((TD & 0xFFFFu) << 16));   // td0 hi | td1 lo
    g1[3] = (int)((TD >> 16) | (128u << 16));     // td1 hi | tile_dim0 = 128
    g1[4] = 64;                                   // tile_dim1 = 64, tile_dim2 = 0
    g1[5] = stride_k;                             // tensor_dim0_stride (elements)
    g1[6] = 0;
    g1[7] = 0;

    // 6-arg form (clang-23 / therock headers): groups 2,3 disabled, cpol = 0
    __builtin_amdgcn_tensor_load_to_lds(g0, g1, (i32x4)0, (i32x4)0, (i32x8)0, 0);
}
#endif

// ---------------------------------------------------------------------------
// Batched left-multiply pass:
//   Out[i, m, n] = sum_k A[m,k] * In[i, k, n]   (complex A; In real or complex)
// element address = i*stride_i + {k|m}*stride_k + n, n contiguous.
// grid = (32, 64): blockIdx.y = batch i, blockIdx.x = 128-column slab.
// Block: 8 waves; the 64x128 input slab is staged into LDS by the TDM
// (padded rows -> bank-conflict-free B-fragment reads), then 16 WMMA tiles
// of 16(m) x 32(n) are computed (2 tasks per wave).
// ---------------------------------------------------------------------------
template<bool IN_COMPLEX, bool OUT_INTERLEAVED>
__global__ __launch_bounds__(256)
void qpool_pass_kernel(const float* __restrict__ in_re,
                       const float* __restrict__ in_im,
                       float* __restrict__ out_re,
                       float* __restrict__ out_im,
                       const float* __restrict__ Gre,
                       const float* __restrict__ Gim,
                       int stride_i, int stride_k)
{
    constexpr int NSLAB = 128;         // columns staged per block
    constexpr int LROW  = NSLAB + 4;   // padded LDS row stride (floats)

    __shared__ float sAre[64 * 64];
    __shared__ float sAim[64 * 64];
    __shared__ float sB[(IN_COMPLEX ? 2 : 1) * 64 * LROW];

    float* sBre = sB;
    float* sBim = sB + 64 * LROW;      // valid only if IN_COMPLEX

    const int i    = blockIdx.y;
    const int nb   = blockIdx.x * NSLAB;
    const int lane = threadIdx.x & 31;
    const int wave = threadIdx.x >> 5;
    const int base_i = i * stride_i;

    // Factor matrix -> LDS (cooperative)
    for (int t = threadIdx.x; t < 64 * 64; t += 256) {
        sAre[t] = Gre[t];
        sAim[t] = Gim[t];
    }

    // Input slab -> LDS
#if QPOOL_HAVE_TDM
    if (wave == 0) {
        tdm_load_slab(in_re + base_i + nb, (unsigned)(uintptr_t)(void*)sBre, stride_k);
        if (IN_COMPLEX)
            tdm_load_slab(in_im + base_i + nb, (unsigned)(uintptr_t)(void*)sBim, stride_k);
        __builtin_amdgcn_s_wait_tensorcnt(0);
    }
#else
    for (int t = threadIdx.x; t < 64 * NSLAB; t += 256) {
        const int k = t >> 7, n = t & (NSLAB - 1);
        sBre[k * LROW + n] = in_re[base_i + k * stride_k + nb + n];
        if (IN_COMPLEX)
            sBim[k * LROW + n] = in_im[base_i + k * stride_k + nb + n];
    }
#endif
    __syncthreads();

    // A-fragment (16x4): lanes 0-15 rows m0..m0+15 K={k0,k0+1}; lanes 16-31 K={k0+2,k0+3}
    const int arow = lane & 15;
    const int akk  = (lane >> 4) * 2;
    // B-fragment (4x16): lane gives column; half-wave selects K pair
    const int bn   = lane & 15;
    const int bk   = (lane >> 4) * 2;
    // C/D (16x16): VGPR j -> row m0 + j + 8*(lane>>4), col n0 + (lane&15)
    const int chi  = (lane >> 4) * 8;

    for (int task = wave; task < 16; task += 8) {
        const int m0 = (task & 3) * 16;
        const int nl = (task >> 2) * 32;    // column offset inside the slab

        v8f acc_re0 = {}; v8f acc_im0 = {};
        v8f acc_re1 = {}; v8f acc_im1 = {};

#pragma unroll
        for (int k0 = 0; k0 < 64; k0 += 4) {
            const int aidx = (m0 + arow) * 64 + k0 + akk;
            v2f ar = *(const v2f*)&sAre[aidx];
            v2f ai = *(const v2f*)&sAim[aidx];
            v2f nai = -ai;

            const int r0 = (k0 + bk) * LROW + nl + bn;
            const int r1 = r0 + LROW;

            v2f br0, br1;
            br0.x = sBre[r0];      br0.y = sBre[r1];
            br1.x = sBre[r0 + 16]; br1.y = sBre[r1 + 16];

            acc_re0 = cwmma(ar, br0, acc_re0);
            acc_im0 = cwmma(ai, br0, acc_im0);
            acc_re1 = cwmma(ar, br1, acc_re1);
            acc_im1 = cwmma(ai, br1, acc_im1);

            if (IN_COMPLEX) {
                v2f bi0, bi1;
                bi0.x = sBim[r0];      bi0.y = sBim[r1];
                bi1.x = sBim[r0 + 16]; bi1.y = sBim[r1 + 16];

                acc_re0 = cwmma(nai, bi0, acc_re0);
                acc_im0 = cwmma(ar,  bi0, acc_im0);
                acc_re1 = cwmma(nai, bi1, acc_re1);
                acc_im1 = cwmma(ar,  bi1, acc_im1);
            }
        }

#pragma unroll
        for (int j = 0; j < 8; ++j) {
            const int m  = m0 + j + chi;
            const int o0 = base_i + m * stride_k + nb + nl + (lane & 15);
            if (OUT_INTERLEAVED) {
                float2* o2 = (float2*)out_re;
                float2 v0; v0.x = acc_re0[j]; v0.y = acc_im0[j];
                float2 v1; v1.x = acc_re1[j]; v1.y = acc_im1[j];
                o2[o0]      = v0;
                o2[o0 + 16] = v1;
            } else {
                out_re[o0]      = acc_re0[j];
                out_im[o0]      = acc_im0[j];
                out_re[o0 + 16] = acc_re1[j];
                out_im[o0 + 16] = acc_im1[j];
            }
        }
    }
}

// ---------------------------------------------------------------------------
// Host launcher
//   out = E x E^H = L( (L(x^T))^H ) with L(M) = (A kron A) M
// ---------------------------------------------------------------------------
extern "C" void kernel_launch(void* const* d_in, const int* in_sizes, int n_in,
                              void* d_out, int out_size, void* d_ws, size_t ws_size,
                              hipStream_t stream)
{
    (void)in_sizes; (void)n_in; (void)out_size; (void)ws_size;

    const float* x = (const float*)d_in[0];   // 4096x4096 f32
    const float* w = (const float*)d_in[1];   // 6 angles

    float* out = (float*)d_out;               // 4096x4096 complex64 (interleaved)
    float* ws  = (float*)d_ws;

    float* Are = ws;
    float* Aim = ws + 64 * 64;
    float* P0  = ws + 2 * 64 * 64;            // 64 MB plane
    float* P1  = P0 + PLANE;                  // 64 MB plane
    float* D0  = out;                         // d_out doubles as scratch planes
    float* D1  = out + PLANE;

    const int S_LO = DIM;        // stride of low base-64 digit (4096)
    const int S_HI = 64 * DIM;   // stride of high base-64 digit (262144)

    const dim3 gT(128, 128), gP(32, 64);

    // K0: factor matrix A (64x64 complex)
    build_factor_kernel<<<1, 1, 0, stream>>>(w, Are, Aim);

    // K1: P0 = x^T  (real)
    transpose_real_kernel<<<gT, 256, 0, stream>>>(x, P0);

    // K2: stage 1 of L on x^T (contract low digit): real -> complex (D0,D1)
    qpool_pass_kernel<false, false><<<gP, 256, 0, stream>>>(
        P0, nullptr, D0, D1, Are, Aim, S_HI, S_LO);

    // K3: stage 2 (contract high digit): Y = E x^T in (P0,P1)
    qpool_pass_kernel<true, false><<<gP, 256, 0, stream>>>(
        D0, D1, P0, P1, Are, Aim, S_LO, S_HI);

    // K4: Q = Y^H  -> (D0,D1)
    transpose_conj_kernel<<<gT, 256, 0, stream>>>(P0, P1, D0, D1);

    // K5: stage 1 of L on Q -> (P0,P1)
    qpool_pass_kernel<true, false><<<gP, 256, 0, stream>>>(
        D0, D1, P0, P1, Are, Aim, S_HI, S_LO);

    // K6: stage 2, write final result interleaved into d_out
    qpool_pass_kernel<true, true><<<gP, 256, 0, stream>>>(
        P0, P1, out, nullptr, Are, Aim, S_LO, S_HI);
}